// ChannelMixer_40200893890845
// MI455X (gfx1250) — compile-verified
//
#include <hip/hip_runtime.h>
#include <hip/hip_bf16.h>

typedef __attribute__((ext_vector_type(2))) float v2f;
typedef __attribute__((ext_vector_type(4))) float v4f;
typedef __attribute__((ext_vector_type(8))) float v8f;

#define LN_EPS 1e-5f
#define DD 1024
#define FF 4096

// ---------------------------------------------------------------------------
// Kernel 1: LayerNorm + time-mix (one workgroup of 256 = 8 waves)
// writes xn to d_out[1024..2048), xk/xr to workspace
// ---------------------------------------------------------------------------
__global__ __launch_bounds__(256) void cm_prelude(
    const float* __restrict__ x, const float* __restrict__ state,
    const float* __restrict__ tmk, const float* __restrict__ tmr,
    const float* __restrict__ lnw, const float* __restrict__ lnb,
    float* __restrict__ xn_out, float* __restrict__ xk, float* __restrict__ xr)
{
    __shared__ float s1[256];
    __shared__ float s2[256];
    const int t = threadIdx.x;

    v4f xv = *(const v4f*)(x + t * 4);
    float s = xv[0] + xv[1] + xv[2] + xv[3];
    float q = xv[0]*xv[0] + xv[1]*xv[1] + xv[2]*xv[2] + xv[3]*xv[3];
    s1[t] = s; s2[t] = q;
    __syncthreads();
    for (int off = 128; off > 0; off >>= 1) {
        if (t < off) { s1[t] += s1[t + off]; s2[t] += s2[t + off]; }
        __syncthreads();
    }
    const float mu   = s1[0] * (1.0f / DD);
    const float var  = s2[0] * (1.0f / DD) - mu * mu;
    const float rstd = rsqrtf(var + LN_EPS);

    v4f w  = *(const v4f*)(lnw   + t * 4);
    v4f b  = *(const v4f*)(lnb   + t * 4);
    v4f pk = *(const v4f*)(tmk   + t * 4);
    v4f pr = *(const v4f*)(tmr   + t * 4);
    v4f pv = *(const v4f*)(state + t * 4);

    v4f xnv, xkv, xrv;
    #pragma unroll
    for (int j = 0; j < 4; ++j) {
        float xn = (xv[j] - mu) * rstd * w[j] + b[j];
        xnv[j] = xn;
        xkv[j] = xn * pk[j] + pv[j] * (1.0f - pk[j]);
        xrv[j] = xn * pr[j] + pv[j] * (1.0f - pr[j]);
    }
    *(v4f*)(xn_out + t * 4) = xnv;
    *(v4f*)(xk     + t * 4) = xkv;
    *(v4f*)(xr     + t * 4) = xrv;
}

// ---------------------------------------------------------------------------
// Kernel 2: k = relu(kw @ xk)^2 (4096 rows) and r = sigmoid(rw @ xr) (1024).
// One 16-row tile per BLOCK; 8 waves split K=1024 into 128 each via
// V_WMMA_F32_16X16X4_F32; deterministic LDS reduction across waves.
// Blocks 0..255 -> kw tiles, 256..319 -> rw tiles.  Grid = 320 blocks.
// ---------------------------------------------------------------------------
__global__ __launch_bounds__(256) void cm_gemv_kr(
    const float* __restrict__ kw, const float* __restrict__ rw,
    const float* __restrict__ xk, const float* __restrict__ xr,
    float* __restrict__ kvec, float* __restrict__ rvec)
{
    __shared__ float vsh[DD];
    __shared__ float red[8][16];
    const int  t    = threadIdx.x;
    const bool is_r = (blockIdx.x >= 256);

    // stage the shared 1024-float vector into LDS (coalesced b128)
    const float* vsrc = is_r ? xr : xk;
    *(v4f*)(vsh + t * 4) = *(const v4f*)(vsrc + t * 4);
    __syncthreads();

    const int wave = t >> 5;
    const int lane = t & 31;
    const int tile = is_r ? (int)blockIdx.x - 256 : (int)blockIdx.x;
    const int m0   = tile * 16;
    const int k0   = wave * (DD / 8);           // split-K: 128 per wave
    const int kofs = (lane >> 4) << 1;          // K offset for this half-wave

    const float* __restrict__ A  = is_r ? rw : kw;
    const float* __restrict__ pa = A + (size_t)(m0 + (lane & 15)) * DD + k0 + kofs;
    const float* __restrict__ pb = vsh + k0 + kofs;

    v8f acc = {};
    #pragma unroll 8
    for (int k = 0; k < DD / 8; k += 4) {
        v2f a = *(const v2f*)(pa + k);          // global_load_b64 (weights)
        v2f b = *(const v2f*)(pb + k);          // ds_load_b64 (vector, bcast)
        acc = __builtin_amdgcn_wmma_f32_16x16x4_f32(
            false, a, false, b, (short)0, acc, false, false);
    }

    // column 0 of D: lane 0 holds rows m0..m0+7, lane 16 holds m0+8..m0+15
    if (lane == 0) {
        #pragma unroll
        for (int j = 0; j < 8; ++j) red[wave][j] = acc[j];
    } else if (lane == 16) {
        #pragma unroll
        for (int j = 0; j < 8; ++j) red[wave][8 + j] = acc[j];
    }
    __syncthreads();

    if (t < 16) {
        float v = 0.0f;
        #pragma unroll
        for (int w = 0; w < 8; ++w) v += red[w][t];
        if (is_r) {
            rvec[m0 + t] = 1.0f / (1.0f + __expf(-v));
        } else {
            float rl = fmaxf(v, 0.0f);
            kvec[m0 + t] = rl * rl;             // squared ReLU
        }
    }
}

// ---------------------------------------------------------------------------
// Kernel 3: partial sums of vw @ k.  vw is 1024x4096.
// Grid = 64 tiles x 4 K-splits = 256 blocks.  Each block: 16-row tile,
// K-chunk of 1024 split 8 ways across waves (128 each), LDS reduce,
// writes 16 partials to ws.  No atomics -> bitwise deterministic.
// ---------------------------------------------------------------------------
__global__ __launch_bounds__(256) void cm_gemv_v_part(
    const float* __restrict__ vw, const float* __restrict__ kvec,
    float* __restrict__ part)
{
    __shared__ float ksh[DD];                   // 4KB chunk of k-vector
    __shared__ float red[8][16];
    const int t     = threadIdx.x;
    const int tile  = blockIdx.x >> 2;          // 0..63
    const int split = blockIdx.x & 3;           // 0..3
    const int kbase = split * DD;               // K-chunk start in [0,4096)

    *(v4f*)(ksh + t * 4) = *(const v4f*)(kvec + kbase + t * 4);
    __syncthreads();

    const int wave = t >> 5;
    const int lane = t & 31;
    const int m0   = tile * 16;
    const int k0   = wave * (DD / 8);           // 128 K per wave
    const int kofs = (lane >> 4) << 1;

    const float* __restrict__ pa =
        vw + (size_t)(m0 + (lane & 15)) * FF + kbase + k0 + kofs;
    const float* __restrict__ pb = ksh + k0 + kofs;

    v8f acc = {};
    #pragma unroll 8
    for (int k = 0; k < DD / 8; k += 4) {
        v2f a = *(const v2f*)(pa + k);
        v2f b = *(const v2f*)(pb + k);
        acc = __builtin_amdgcn_wmma_f32_16x16x4_f32(
            false, a, false, b, (short)0, acc, false, false);
    }

    if (lane == 0) {
        #pragma unroll
        for (int j = 0; j < 8; ++j) red[wave][j] = acc[j];
    } else if (lane == 16) {
        #pragma unroll
        for (int j = 0; j < 8; ++j) red[wave][8 + j] = acc[j];
    }
    __syncthreads();

    if (t < 16) {
        float s = 0.0f;
        #pragma unroll
        for (int w = 0; w < 8; ++w) s += red[w][t];
        part[blockIdx.x * 16 + t] = s;          // [(tile*4+split)*16 + row]
    }
}

// ---------------------------------------------------------------------------
// Kernel 4: out[row] = x[row] + r[row] * sum_{split<4} part[...]
// Fixed-order reduction -> deterministic.  Grid = 4 x 256 (one row/thread).
// ---------------------------------------------------------------------------
__global__ __launch_bounds__(256) void cm_epilogue(
    const float* __restrict__ part, const float* __restrict__ rvec,
    const float* __restrict__ x, float* __restrict__ out)
{
    const int row  = blockIdx.x * 256 + threadIdx.x;   // 0..1023
    const int tile = row >> 4;
    const int r    = row & 15;
    float s = 0.0f;
    #pragma unroll
    for (int split = 0; split < 4; ++split)
        s += part[(tile * 4 + split) * 16 + r];
    out[row] = x[row] + rvec[row] * s;
}

// ---------------------------------------------------------------------------
extern "C" void kernel_launch(void* const* d_in, const int* in_sizes, int n_in,
                              void* d_out, int out_size, void* d_ws, size_t ws_size,
                              hipStream_t stream) {
    const float* x     = (const float*)d_in[0];
    const float* state = (const float*)d_in[1];
    const float* tmk   = (const float*)d_in[2];
    const float* tmr   = (const float*)d_in[3];
    const float* kw    = (const float*)d_in[4];
    const float* vw    = (const float*)d_in[5];
    const float* rw    = (const float*)d_in[6];
    const float* lnw   = (const float*)d_in[7];
    const float* lnb   = (const float*)d_in[8];

    float* out = (float*)d_out;          // [out(1024) | xn(1024)]
    float* ws  = (float*)d_ws;
    float* xk   = ws;                    // 1024
    float* xr   = ws + 1024;             // 1024
    float* rvec = ws + 2048;             // 1024
    float* kvec = ws + 3072;             // 4096
    float* part = ws + 7168;             // 4096 (64 tiles * 4 splits * 16 rows)

    cm_prelude    <<<1,   256, 0, stream>>>(x, state, tmk, tmr, lnw, lnb,
                                            out + DD, xk, xr);
    cm_gemv_kr    <<<320, 256, 0, stream>>>(kw, rw, xk, xr, kvec, rvec);
    cm_gemv_v_part<<<256, 256, 0, stream>>>(vw, kvec, part);
    cm_epilogue   <<<4,   256, 0, stream>>>(part, rvec, x, out);
}